// CrossCorrelationLoss_71201967834017
// MI455X (gfx1250) — compile-verified
//
#include <hip/hip_runtime.h>

// ---------------------------------------------------------------------------
// Fused 3D local NCC loss for MI455X (gfx1250, wave32).
// All three separable 9-tap box-filter passes are done as banded-matrix GEMMs
// with V_WMMA_F32_16X16X4_F32 (exact: 0/1 weights, f32 accumulate).
// One workgroup = one 16x16x16 output tile; halo tiles staged in LDS.
// ---------------------------------------------------------------------------

namespace {
constexpr int D_DIM = 160, H_DIM = 192, W_DIM = 160, BATCH = 2;
constexpr int TILE = 16;          // output tile edge (matches WMMA M/N)
constexpr int HALO = 24;          // TILE + 2*4 halo
constexpr int PADR = 4;
constexpr int TX = W_DIM / TILE;  // 10
constexpr int TY = H_DIM / TILE;  // 12
constexpr int TZ = D_DIM / TILE;  // 10
constexpr int BLOCKS_PER_BATCH = TX * TY * TZ;     // 1200
constexpr int NBLOCKS = BATCH * BLOCKS_PER_BATCH;  // 2400
constexpr float WIN3F = 729.0f;
constexpr float INV_WIN3 = 1.0f / 729.0f;
}

typedef __attribute__((ext_vector_type(2))) float v2f;
typedef __attribute__((ext_vector_type(8))) float v8f;

// D = A(16x4) * B(4x16) + C, f32 WMMA (8-arg pattern like the f16 variants)
__device__ __forceinline__ v8f wmma4(v2f a, v2f b, v8f c) {
  return __builtin_amdgcn_wmma_f32_16x16x4_f32(
      /*neg_a=*/false, a, /*neg_b=*/false, b,
      /*c_mod=*/(short)0, c, /*reuse_a=*/false, /*reuse_b=*/false);
}

// Banded box-filter matrix slice: A[m][k] = 1 iff m <= k <= m+8
__device__ __forceinline__ v2f band(int k, int mm) {
  v2f a;
  a.x = (k >= mm && k <= mm + 8) ? 1.0f : 0.0f;
  a.y = (k + 1 >= mm && k + 1 <= mm + 8) ? 1.0f : 0.0f;
  return a;
}

__global__ __launch_bounds__(256) void ncc_main(const float* __restrict__ gI,
                                                const float* __restrict__ gJ,
                                                float* __restrict__ partial) {
  // LDS budget: 2*13824*4 (halo I,J) + 6144*4 (one-channel HW sums)
  //           + 8*384*4 (per-wave W-sum scratch) + 32  ~= 147.5 KB
  __shared__ float sI[HALO * HALO * HALO];
  __shared__ float sJ[HALO * HALO * HALO];
  __shared__ float sHW[HALO * TILE * TILE];  // per-channel [d][h_out][w]
  __shared__ float sWs[8][HALO * TILE];      // per-wave   [h][w_out]
  __shared__ float sRed[8];

  const int tid = threadIdx.x;
  const int wave = tid >> 5;
  const int lane = tid & 31;

  int blk = blockIdx.x;
  const int b = blk / BLOCKS_PER_BATCH;
  int r = blk % BLOCKS_PER_BATCH;
  const int tz = r / (TY * TX); r %= (TY * TX);
  const int ty = r / TX;
  const int tx = r % TX;

  const int d0 = tz * TILE - PADR;
  const int h0 = ty * TILE - PADR;
  const int w0 = tx * TILE - PADR;
  const long gbase = (long)b * D_DIM * H_DIM * W_DIM;

  // ---- Stage halo tiles of I and J into LDS (zero padding at borders) ----
  for (int idx = tid; idx < HALO * HALO * HALO; idx += 256) {
    int dz = idx / (HALO * HALO);
    int rem = idx % (HALO * HALO);
    int dy = rem / HALO;
    int dx = rem % HALO;
    int gz = d0 + dz, gy = h0 + dy, gx = w0 + dx;
    float vi = 0.0f, vj = 0.0f;
    if ((unsigned)gz < (unsigned)D_DIM && (unsigned)gy < (unsigned)H_DIM &&
        (unsigned)gx < (unsigned)W_DIM) {
      long g = gbase + ((long)gz * H_DIM + gy) * W_DIM + gx;
      vi = gI[g];
      vj = gJ[g];
    }
    sI[idx] = vi;
    sJ[idx] = vj;
  }
  __syncthreads();

  // Per-lane WMMA operand geometry (ISA §7.12.2 layouts, wave32):
  const int m = lane & 15;               // A: M row / B: N col / D: N col
  const int koff = (lane >> 4) << 1;     // K pair offset within a 4-chunk
  const int mbase = (lane >> 4) << 3;    // D rows held: mbase..mbase+7

  v8f dacc[10];  // [hi*5 + c] : per-channel D-pass tiles kept in registers

  // ---- Channels: 0=I 1=J 2=I*I 3=J*J 4=I*J ----
  for (int c = 0; c < 5; ++c) {
    // Pass 1 (W box-sum) + Pass 2 (H box-sum), per d-slab, 3 slabs per wave
    for (int d = wave; d < HALO; d += 8) {
      const float* slI = &sI[d * HALO * HALO];
      const float* slJ = &sJ[d * HALO * HALO];
      // -- Pass 1: two 16-row h-tiles of Wsum[h][w_out] --
      for (int ht = 0; ht < 2; ++ht) {
        int h = ht * 16 + m;
        if (h > HALO - 1) h = HALO - 1;  // clamped lanes: results discarded
        v8f acc = {};
#pragma unroll
        for (int ch = 0; ch < 6; ++ch) {
          int k = ch * 4 + koff;  // input w index (0..23)
          float i0 = slI[h * HALO + k], i1 = slI[h * HALO + k + 1];
          float j0 = slJ[h * HALO + k], j1 = slJ[h * HALO + k + 1];
          v2f bv;
          if (c == 0)      { bv.x = i0;      bv.y = i1; }
          else if (c == 1) { bv.x = j0;      bv.y = j1; }
          else if (c == 2) { bv.x = i0 * i0; bv.y = i1 * i1; }
          else if (c == 3) { bv.x = j0 * j0; bv.y = j1 * j1; }
          else             { bv.x = i0 * j0; bv.y = i1 * j1; }
          acc = wmma4(band(k, m), bv, acc);
        }
        int hrow = ht * 16 + m;  // D layout: this lane's N index is m
        if (hrow < HALO) {
#pragma unroll
          for (int i = 0; i < 8; ++i)
            sWs[wave][hrow * TILE + (mbase + i)] = acc[i];
        }
      }
      // same-wave LDS store->load ordering before Pass 2 reads sWs
      asm volatile("s_wait_dscnt 0x0" ::: "memory");

      // -- Pass 2: H box-sum. M=h_out, N=w, K=h(24) --
      {
        v8f acc = {};
#pragma unroll
        for (int ch = 0; ch < 6; ++ch) {
          int k = ch * 4 + koff;  // h index
          v2f bv;
          bv.x = sWs[wave][k * TILE + m];
          bv.y = sWs[wave][(k + 1) * TILE + m];
          acc = wmma4(band(k, m), bv, acc);
        }
#pragma unroll
        for (int i = 0; i < 8; ++i)
          sHW[(d * TILE + (mbase + i)) * TILE + m] = acc[i];
      }
    }
    __syncthreads();  // all slabs of channel c visible to all waves

    // -- Pass 3: D box-sum. M=d_out, N=w, K=d(24); 2 h-rows per wave --
#pragma unroll
    for (int hi = 0; hi < 2; ++hi) {
      const int hh = wave + 8 * hi;  // h_out row 0..15
      v8f acc = {};
#pragma unroll
      for (int ch = 0; ch < 6; ++ch) {
        int k = ch * 4 + koff;  // d index
        v2f bv;
        bv.x = sHW[(k * TILE + hh) * TILE + m];
        bv.y = sHW[((k + 1) * TILE + hh) * TILE + m];
        acc = wmma4(band(k, m), bv, acc);
      }
      dacc[hi * 5 + c] = acc;
    }
    __syncthreads();  // sHW reads done before next channel overwrites it
  }

  // ---- Pointwise NCC combine + block reduction ----
  float local = 0.0f;
#pragma unroll
  for (int hi = 0; hi < 2; ++hi) {
#pragma unroll
    for (int i = 0; i < 8; ++i) {
      float Is = dacc[hi * 5 + 0][i];
      float Js = dacc[hi * 5 + 1][i];
      float I2 = dacc[hi * 5 + 2][i];
      float J2 = dacc[hi * 5 + 3][i];
      float IJ = dacc[hi * 5 + 4][i];
      float uI = Is * INV_WIN3;
      float uJ = Js * INV_WIN3;
      float cross = IJ - uJ * Is - uI * Js + uI * uJ * WIN3F;
      float Iv = I2 - 2.0f * uI * Is + uI * uI * WIN3F;
      float Jv = J2 - 2.0f * uJ * Js + uJ * uJ * WIN3F;
      local += (cross * cross) / (Iv * Jv + 1e-5f);
    }
  }
#pragma unroll
  for (int off = 16; off >= 1; off >>= 1)
    local += __shfl_xor(local, off, 32);
  if (lane == 0) sRed[wave] = local;
  __syncthreads();
  if (tid == 0) {
    float s = 0.0f;
#pragma unroll
    for (int i = 0; i < 8; ++i) s += sRed[i];
    partial[blockIdx.x] = s;
  }
}

// Deterministic final mean over the 2400 per-block partials.
__global__ __launch_bounds__(256) void ncc_reduce(const float* __restrict__ partial,
                                                  float* __restrict__ out) {
  __shared__ float red[256];
  float s = 0.0f;
  for (int i = (int)threadIdx.x; i < NBLOCKS; i += 256) s += partial[i];
  red[threadIdx.x] = s;
  __syncthreads();
  for (int off = 128; off >= 1; off >>= 1) {
    if ((int)threadIdx.x < off) red[threadIdx.x] += red[threadIdx.x + off];
    __syncthreads();
  }
  if (threadIdx.x == 0) {
    const double ntot = (double)BATCH * D_DIM * H_DIM * W_DIM;  // 9,830,400
    out[0] = (float)((double)red[0] / ntot);
  }
}

extern "C" void kernel_launch(void* const* d_in, const int* in_sizes, int n_in,
                              void* d_out, int out_size, void* d_ws, size_t ws_size,
                              hipStream_t stream) {
  (void)in_sizes; (void)n_in; (void)out_size; (void)ws_size;
  const float* I = (const float*)d_in[0];  // predicted, fp32
  const float* J = (const float*)d_in[1];  // target,    fp32
  float* out = (float*)d_out;              // scalar fp32
  float* partial = (float*)d_ws;           // NBLOCKS floats of scratch

  ncc_main<<<dim3(NBLOCKS), dim3(256), 0, stream>>>(I, J, partial);
  ncc_reduce<<<dim3(1), dim3(256), 0, stream>>>(partial, out);
}